// NCNPredictor_Sparse_541165879725
// MI455X (gfx1250) — compile-verified
//
#include <hip/hip_runtime.h>

typedef __attribute__((ext_vector_type(2))) float v2f;
typedef __attribute__((ext_vector_type(8))) float v8f;

#define N_NODES  10000
#define N_EDGES  320000
#define N_PAIRS  2048
#define IN_CH    128
#define HIDDEN   512
#define CHUNK    256     // pairs per chunk (bounds scratch to ~27 MB, L2-resident)
#define PADK     10240   // padded row stride for Ai/Aj rows
#define EDGE_TILE 1024
#define KSPLIT   10      // K-slices for cn GEMM (10000 / 10 = 1000, mult of 4)

// ---------------- zero scratch ----------------
__global__ void zero_rows(float* __restrict__ p, int n) {
    int i = blockIdx.x * blockDim.x + threadIdx.x;
    int stride = gridDim.x * blockDim.x;
    for (; i < n; i += stride) p[i] = 0.0f;
}

// ---- scatter symmetrized adjacency rows for one chunk of pairs ----
// blockDim.x == CHUNK; each thread owns one pair, block stages an edge tile in
// LDS via CDNA5 async global->LDS copies (ASYNCcnt).
__global__ void scatter_rows(const int* __restrict__ ei,
                             const int* __restrict__ ti, const int* __restrict__ tj,
                             int pair0,
                             float* __restrict__ Ai, float* __restrict__ Aj) {
    __shared__ __align__(16) int su[EDGE_TILE];
    __shared__ __align__(16) int sv[EDGE_TILE];
    const int p    = threadIdx.x;
    const int myTi = ti[pair0 + p];
    const int myTj = tj[pair0 + p];
    float* rowi = Ai + (size_t)p * PADK;
    float* rowj = Aj + (size_t)p * PADK;

    const int e0   = blockIdx.x * EDGE_TILE;
    const int rem  = N_EDGES - e0;
    const int ecnt = rem < EDGE_TILE ? rem : EDGE_TILE;   // always a multiple of 4

    // async bulk copy: 4 ints (b128) per thread per array, direct to LDS
    const int t4 = threadIdx.x * 4;
    if (t4 < ecnt) {
        unsigned lds_u = (unsigned)(size_t)&su[t4];
        unsigned lds_v = (unsigned)(size_t)&sv[t4];
        const int* gu = ei + e0 + t4;
        const int* gv = ei + N_EDGES + e0 + t4;
        asm volatile("global_load_async_to_lds_b128 %0, %1, off"
                     :: "v"(lds_u), "v"(gu) : "memory");
        asm volatile("global_load_async_to_lds_b128 %0, %1, off"
                     :: "v"(lds_v), "v"(gv) : "memory");
    }
    asm volatile("s_wait_asynccnt 0x0" ::: "memory");
    __syncthreads();

    for (int t = 0; t < ecnt; ++t) {
        const int u = su[t];
        const int v = sv[t];
        // A[u,v] += 1 and A[v,u] += 1 (symmetrize + coalesce-sum)
        if (u == myTi) atomicAdd(&rowi[v], 1.0f);
        if (v == myTi) atomicAdd(&rowi[u], 1.0f);
        if (u == myTj) atomicAdd(&rowj[v], 1.0f);
        if (v == myTj) atomicAdd(&rowj[u], 1.0f);
    }
}

// ---------------- xij = x[ti] * x[tj] into xs[:, 0:128] ----------------
__global__ void pair_xij(const float* __restrict__ x,
                         const int* __restrict__ ti, const int* __restrict__ tj,
                         float* __restrict__ xs) {
    const int b = blockIdx.x;
    const int c = threadIdx.x;   // blockDim == IN_CH
    xs[(size_t)b * (2 * IN_CH) + c] =
        x[(size_t)ti[b] * IN_CH + c] * x[(size_t)tj[b] * IN_CH + c];
}

// ---- cn_emb += (Ai ⊙ Aj) @ x  via V_WMMA_F32_16X16X4_F32 --------------------
// One wave per (M-tile, K-slice); the wave holds all 8 N-tiles (c[8], 64 VGPRs)
// so each Ai/Aj row is read once per K-slice. Partial sums accumulate with
// global_atomic_add_f32 into the pre-zeroed cn columns of xs.
__global__ void cn_gemm(const float* __restrict__ Ai, const float* __restrict__ Aj,
                        const float* __restrict__ x, float* __restrict__ xs, int pair0) {
    const int lane = threadIdx.x;        // 32 lanes, one wave
    const int half = lane >> 4;
    const int m    = lane & 15;
    const int mt   = blockIdx.x * 16;                    // row tile within chunk
    const int k0   = blockIdx.y * (N_NODES / KSPLIT);    // K-slice start
    const int k1   = k0 + (N_NODES / KSPLIT);
    const float* ri = Ai + (size_t)(mt + m) * PADK;
    const float* rj = Aj + (size_t)(mt + m) * PADK;

    v8f c[8];
#pragma unroll
    for (int t = 0; t < 8; ++t) c[t] = v8f{};

    for (int k = k0; k < k1; k += 4) {
        const int ka = k + (half ? 2 : 0);
        // A fragment (16x4): fuse the Hadamard product Ai*Aj
        v2f a;
        a.x = ri[ka]     * rj[ka];
        a.y = ri[ka + 1] * rj[ka + 1];
        const float* xr0 = x + (size_t)ka * IN_CH + m;   // row ka, col m + 16t
        const float* xr1 = xr0 + IN_CH;                  // row ka+1
#pragma unroll
        for (int t = 0; t < 8; ++t) {
            v2f b;
            b.x = xr0[16 * t];
            b.y = xr1[16 * t];
            c[t] = __builtin_amdgcn_wmma_f32_16x16x4_f32(false, a, false, b,
                                                         (short)0, c[t], false, false);
        }
    }
    // C layout: VGPR r holds row r (lanes 0-15) / row 8+r (lanes 16-31), col = lane&15
#pragma unroll
    for (int t = 0; t < 8; ++t) {
        const int n0 = 16 * t;
#pragma unroll
        for (int r = 0; r < 8; ++r) {
            const int row = (half ? 8 : 0) + r;
            atomicAdd(&xs[(size_t)(pair0 + mt + row) * (2 * IN_CH) + IN_CH + n0 + m],
                      c[t][r]);
        }
    }
}

// ---------------- h = relu(xs @ W1 + b1) via WMMA f32 ----------------
__global__ void mlp_gemm(const float* __restrict__ xs, const float* __restrict__ W1,
                         const float* __restrict__ b1, float* __restrict__ h) {
    const int lane = threadIdx.x;
    const int half = lane >> 4;
    const int m    = lane & 15;
    const int m0   = blockIdx.x * 16;    // over N_PAIRS
    const int n0   = blockIdx.y * 16;    // over HIDDEN

    const float* arow = xs + (size_t)(m0 + m) * (2 * IN_CH);
    v8f c = {};
    for (int k = 0; k < 2 * IN_CH; k += 4) {
        const int ka = k + (half ? 2 : 0);
        v2f a, b;
        a.x = arow[ka];
        a.y = arow[ka + 1];
        b.x = W1[(size_t)ka       * HIDDEN + n0 + m];
        b.y = W1[(size_t)(ka + 1) * HIDDEN + n0 + m];
        c = __builtin_amdgcn_wmma_f32_16x16x4_f32(false, a, false, b, (short)0, c,
                                                  false, false);
    }
    const float bias = b1[n0 + m];
#pragma unroll
    for (int r = 0; r < 8; ++r) {
        const int row = (half ? 8 : 0) + r;
        const float v = c[r] + bias;
        h[(size_t)(m0 + row) * HIDDEN + n0 + m] = v > 0.0f ? v : 0.0f;
    }
}

// ---------------- out = h @ W2 + b2  (per-pair 512-dot, wave32 reduce) ----------------
__global__ void out_gemv(const float* __restrict__ h, const float* __restrict__ W2,
                         const float* __restrict__ b2, float* __restrict__ out) {
    const int wavesPerBlock = blockDim.x >> 5;
    const int b    = blockIdx.x * wavesPerBlock + (threadIdx.x >> 5);
    const int lane = threadIdx.x & 31;
    float s = 0.0f;
    for (int i = lane; i < HIDDEN; i += 32)
        s += h[(size_t)b * HIDDEN + i] * W2[i];
    for (int off = 16; off > 0; off >>= 1)
        s += __shfl_down(s, off, 32);
    if (lane == 0) out[b] = s + b2[0];
}

extern "C" void kernel_launch(void* const* d_in, const int* in_sizes, int n_in,
                              void* d_out, int out_size, void* d_ws, size_t ws_size,
                              hipStream_t stream) {
    const float* x   = (const float*)d_in[0];
    const int*   ei  = (const int*)d_in[1];   // [2][N_EDGES]
    const int*   tar = (const int*)d_in[2];   // [2][N_PAIRS]
    const float* W1  = (const float*)d_in[3];
    const float* b1  = (const float*)d_in[4];
    const float* W2  = (const float*)d_in[5];
    const float* b2  = (const float*)d_in[6];
    float* out = (float*)d_out;

    const int* ti = tar;
    const int* tj = tar + N_PAIRS;

    // workspace layout (~27 MB total, L2-resident)
    float* Ai = (float*)d_ws;                       // CHUNK * PADK
    float* Aj = Ai + (size_t)CHUNK * PADK;          // CHUNK * PADK
    float* xs = Aj + (size_t)CHUNK * PADK;          // N_PAIRS * 256
    float* h  = xs + (size_t)N_PAIRS * 2 * IN_CH;   // N_PAIRS * 512

    // zero xs (cn columns accumulate atomically), then fill xij half
    zero_rows<<<512, 256, 0, stream>>>(xs, N_PAIRS * 2 * IN_CH);
    pair_xij<<<N_PAIRS, IN_CH, 0, stream>>>(x, ti, tj, xs);

    const int nScatterBlocks = (N_EDGES + EDGE_TILE - 1) / EDGE_TILE;
    for (int pair0 = 0; pair0 < N_PAIRS; pair0 += CHUNK) {
        // zero both Ai and Aj (contiguous)
        zero_rows<<<1024, 256, 0, stream>>>(Ai, 2 * CHUNK * PADK);
        scatter_rows<<<nScatterBlocks, CHUNK, 0, stream>>>(ei, ti, tj, pair0, Ai, Aj);
        dim3 g(CHUNK / 16, KSPLIT);
        cn_gemm<<<g, 32, 0, stream>>>(Ai, Aj, x, xs, pair0);
    }

    dim3 g2(N_PAIRS / 16, HIDDEN / 16);
    mlp_gemm<<<g2, 32, 0, stream>>>(xs, W1, b1, h);

    out_gemv<<<N_PAIRS / 8, 256, 0, stream>>>(h, W2, b2, out);
}